// SupConLoss_33174327394581
// MI455X (gfx1250) — compile-verified
//
#include <hip/hip_runtime.h>
#include <math.h>

// ---------------------------------------------------------------------------
// SupConLoss on MI455X (gfx1250, wave32).
// features: [B=4096, V=2, D=128] f32; labels: [4096] int.  Output: scalar f32.
//
// Pipeline (all on `stream`):
//   1) normalize_kernel : Q[n] = cf[n]/||cf[n]||  (cf = view-major reorder)
//   2) zero_kernel      : S[0..N-1] = 0, accum = 0
//   3) colsum_kernel    : S[j] += exp(logits[i][j]) * neg_mask   (WMMA f32)
//   4) loss_kernel      : accum += pos_mask * (l - log(S[j]+exp(l)))  (WMMA f32)
//   5) finalize_kernel  : out = -accum / N
// Logits are recomputed per pass via V_WMMA_F32_16X16X4_F32; the 256 MB
// logits matrix is never materialized. Q (4 MB) stays resident in L2.
// ---------------------------------------------------------------------------

typedef __attribute__((ext_vector_type(2))) float v2f;
typedef __attribute__((ext_vector_type(8))) float v8f;

#define B_   4096
#define N_   8192
#define D_   128
#define NTI  (N_ / 16)     // 512 row tiles
#define TJW  4             // j-tiles per wave (16x64 strip)
#define NTJG (N_ / (16 * TJW))  // 128 j-tile groups

__constant__ float kInvT = 1.0f / 0.07f;   // 1/temperature

// ---------------------------------------------------------------------------
// 1) Row-normalize into workspace Q.  One block (128 threads) per row.
// ---------------------------------------------------------------------------
__global__ __launch_bounds__(128) void normalize_kernel(
    const float* __restrict__ feat, float* __restrict__ Q) {
  const int n = blockIdx.x;          // 0..N-1 (view-major: n = v*B + b)
  const int v = n >> 12;             // n / B_
  const int b = n & (B_ - 1);        // n % B_
  const int t = threadIdx.x;         // 0..127
  const float x = feat[((size_t)b * 2 + v) * D_ + t];

  __shared__ float red[128];
  red[t] = x * x;
  __syncthreads();
#pragma unroll
  for (int s = 64; s > 0; s >>= 1) {
    if (t < s) red[t] += red[t + s];
    __syncthreads();
  }
  const float inv = rsqrtf(red[0]);
  Q[(size_t)n * D_ + t] = x * inv;
}

// ---------------------------------------------------------------------------
// 2) Zero S[N_] and the scalar accumulator (workspace is poisoned by harness).
// ---------------------------------------------------------------------------
__global__ void zero_kernel(float* __restrict__ p, int count) {
  const int i = blockIdx.x * blockDim.x + threadIdx.x;
  if (i < count) p[i] = 0.0f;
}

// ---------------------------------------------------------------------------
// Per-wave 16x64 logits strip via V_WMMA_F32_16X16X4_F32.
//   A (16x4 f32): lanes 0-15 hold M=lane, K=kb..kb+1; lanes 16-31 K=kb+2..kb+3
//   B (4x16 f32): same per-lane addressing with N=lane&15 (rows of Q[j]).
//   C/D (16x16 f32): lane -> column n=lane&15; VGPR r -> row 8*(lane>>4)+r.
// ---------------------------------------------------------------------------
__device__ __forceinline__ void wave_logits_strip(
    const float* __restrict__ Q, int i0, int j0, v8f acc[TJW]) {
  const int lane = threadIdx.x & 31;
  const int n    = lane & 15;
  const int kb   = (lane >> 4) << 1;          // 0 or 2

  const float* arow = Q + (size_t)(i0 + n) * D_ + kb;
  const float* brow0 = Q + (size_t)(j0 +  0 + n) * D_ + kb;
  const float* brow1 = Q + (size_t)(j0 + 16 + n) * D_ + kb;
  const float* brow2 = Q + (size_t)(j0 + 32 + n) * D_ + kb;
  const float* brow3 = Q + (size_t)(j0 + 48 + n) * D_ + kb;

#pragma unroll
  for (int t = 0; t < TJW; ++t) acc[t] = (v8f){};

#pragma unroll
  for (int k0 = 0; k0 < D_; k0 += 4) {
    const v2f a  = *(const v2f*)(arow + k0);
    const v2f b0 = *(const v2f*)(brow0 + k0);
    const v2f b1 = *(const v2f*)(brow1 + k0);
    const v2f b2 = *(const v2f*)(brow2 + k0);
    const v2f b3 = *(const v2f*)(brow3 + k0);
    acc[0] = __builtin_amdgcn_wmma_f32_16x16x4_f32(false, a, false, b0,
                                                   (short)0, acc[0], false, false);
    acc[1] = __builtin_amdgcn_wmma_f32_16x16x4_f32(false, a, false, b1,
                                                   (short)0, acc[1], false, false);
    acc[2] = __builtin_amdgcn_wmma_f32_16x16x4_f32(false, a, false, b2,
                                                   (short)0, acc[2], false, false);
    acc[3] = __builtin_amdgcn_wmma_f32_16x16x4_f32(false, a, false, b3,
                                                   (short)0, acc[3], false, false);
  }
  const float s = kInvT;
#pragma unroll
  for (int t = 0; t < TJW; ++t) {
#pragma unroll
    for (int r = 0; r < 8; ++r) acc[t][r] *= s;
  }
}

// ---------------------------------------------------------------------------
// 3) S[j] = sum_i exp(logits[i][j]) * (labels[i%B] != labels[j%B])
// ---------------------------------------------------------------------------
__global__ __launch_bounds__(256) void colsum_kernel(
    const float* __restrict__ Q, const int* __restrict__ labels,
    float* __restrict__ S) {
  const int wave = (blockIdx.x * blockDim.x + threadIdx.x) >> 5;
  const int ti   = wave >> 7;          // / NTJG
  const int tjg  = wave & (NTJG - 1);
  const int i0   = ti * 16;
  const int j0   = tjg * (16 * TJW);

  v8f acc[TJW];
  wave_logits_strip(Q, i0, j0, acc);

  const int lane  = threadIdx.x & 31;
  const int n     = lane & 15;
  const int mbase = (lane >> 4) * 8;

#pragma unroll
  for (int t = 0; t < TJW; ++t) {
    const int j    = j0 + t * 16 + n;
    const int labj = labels[j & (B_ - 1)];
    float partial = 0.0f;
#pragma unroll
    for (int r = 0; r < 8; ++r) {
      const int i    = i0 + mbase + r;
      const int labi = labels[i & (B_ - 1)];
      partial += (labi != labj) ? __expf(acc[t][r]) : 0.0f;
    }
    partial += __shfl_xor(partial, 16, 32);   // fold the two half-wave rows
    if (lane < 16) atomicAdd(&S[j], partial);
  }
}

// ---------------------------------------------------------------------------
// 4) accum += sum over (i,j): pos_mask * (l - log(S[j] + exp(l)))
//    pos_mask = (labels equal) && (i != j)
// ---------------------------------------------------------------------------
__global__ __launch_bounds__(256) void loss_kernel(
    const float* __restrict__ Q, const int* __restrict__ labels,
    const float* __restrict__ S, float* __restrict__ accum) {
  const int wave = (blockIdx.x * blockDim.x + threadIdx.x) >> 5;
  const int ti   = wave >> 7;
  const int tjg  = wave & (NTJG - 1);
  const int i0   = ti * 16;
  const int j0   = tjg * (16 * TJW);

  v8f acc[TJW];
  wave_logits_strip(Q, i0, j0, acc);

  const int lane  = threadIdx.x & 31;
  const int n     = lane & 15;
  const int mbase = (lane >> 4) * 8;

  float part = 0.0f;
#pragma unroll
  for (int t = 0; t < TJW; ++t) {
    const int j    = j0 + t * 16 + n;
    const int labj = labels[j & (B_ - 1)];
    const float Sj = S[j];
#pragma unroll
    for (int r = 0; r < 8; ++r) {
      const int i    = i0 + mbase + r;
      const int labi = labels[i & (B_ - 1)];
      if (labi == labj && i != j) {
        const float l = acc[t][r];
        part += l - __logf(Sj + __expf(l));
      }
    }
  }

  // block-level reduction, then one atomic per block
  __shared__ float red[256];
  const int tid = threadIdx.x;
  red[tid] = part;
  __syncthreads();
#pragma unroll
  for (int s = 128; s > 0; s >>= 1) {
    if (tid < s) red[tid] += red[tid + s];
    __syncthreads();
  }
  if (tid == 0) atomicAdd(accum, red[0]);
}

// ---------------------------------------------------------------------------
// 5) out = -(T/Tbase) * accum / N   (T/Tbase == 1)
// ---------------------------------------------------------------------------
__global__ void finalize_kernel(const float* __restrict__ accum,
                                float* __restrict__ out) {
  out[0] = -accum[0] / (float)N_;
}

extern "C" void kernel_launch(void* const* d_in, const int* in_sizes, int n_in,
                              void* d_out, int out_size, void* d_ws, size_t ws_size,
                              hipStream_t stream) {
  const float* feat   = (const float*)d_in[0];   // [4096, 2, 128]
  const int*   labels = (const int*)d_in[1];     // [4096]
  float*       out    = (float*)d_out;           // [1]

  float* Q     = (float*)d_ws;                   // N_ * D_ floats (4 MB)
  float* S     = Q + (size_t)N_ * D_;            // N_ floats
  float* accum = S + N_;                         // 1 float

  // 1) normalize rows of cf into Q
  normalize_kernel<<<N_, 128, 0, stream>>>(feat, Q);

  // 2) zero S and accum
  zero_kernel<<<(N_ + 1 + 255) / 256, 256, 0, stream>>>(S, N_ + 1);

  // 3)+4) two WMMA passes over the 8192x8192 logits, 16x64 strip per wave
  const int total_waves = NTI * NTJG;            // 512 * 128 = 65536
  const int blocks = total_waves / 8;            // 8 waves per 256-thread block
  colsum_kernel<<<blocks, 256, 0, stream>>>(Q, labels, S);
  loss_kernel<<<blocks, 256, 0, stream>>>(Q, labels, S, accum);

  // 5) scalar finalize
  finalize_kernel<<<1, 1, 0, stream>>>(accum, out);
}